// HeterogeneousGNN_52080773432123
// MI455X (gfx1250) — compile-verified
//
#include <hip/hip_runtime.h>

// MI455X / gfx1250 implementation of the 2-layer heterogeneous GNN.
//
// Roofline: ~1.6 GFLOP total vs ~154 MB of HBM edge traffic and ~6.6 GB of
// L2-resident gather/scatter traffic (all embeddings + accumulators fit in the
// 192 MB L2). The kernel set is therefore organized around L2 atomic-add
// throughput: one wave32 per edge, float2 per lane (64-dim rows), scalar
// edge broadcast via readfirstlane, global_atomic_add_f32 scatters.
// The final mean (x * 1/3) is done with exact f32 WMMA (V_WMMA_F32_16X16X4_F32)
// as D = (c*I16) x X over 16x16 tiles, chained over four K=4 slices.

typedef __attribute__((ext_vector_type(2))) float v2f;
typedef __attribute__((ext_vector_type(8))) float v8f;

#define DIM 64

// ---------------------------------------------------------------------------
// Init: cur = sum = embedding ; acc = 0.   float4-vectorized.
// ---------------------------------------------------------------------------
__global__ void __launch_bounds__(256) gnn_init(
    const float4* __restrict__ ue, const float4* __restrict__ ie,
    float4* __restrict__ u_cur, float4* __restrict__ i_cur,
    float4* __restrict__ u_acc, float4* __restrict__ i_acc,
    float4* __restrict__ u_sum, float4* __restrict__ i_sum,
    int nu4, int ni4)
{
    int idx = blockIdx.x * blockDim.x + threadIdx.x;
    const float4 z = make_float4(0.f, 0.f, 0.f, 0.f);
    if (idx < nu4) {
        float4 e = ue[idx];
        u_cur[idx] = e; u_sum[idx] = e; u_acc[idx] = z;
    } else {
        int j = idx - nu4;
        if (j < ni4) {
            float4 e = ie[j];
            i_cur[j] = e; i_sum[j] = e; i_acc[j] = z;
        }
    }
}

// ---------------------------------------------------------------------------
// COO SpMM scatter: acc[rows[e], :] += vals[e] * x[cols[e], :]
// One wave32 per edge; lane handles dims {2*lane, 2*lane+1} (one B64 load,
// two global_atomic_add_f32 that resolve in L2). Edge triple is broadcast to
// SGPRs via readfirstlane so the feature loads use the saddr form.
// ---------------------------------------------------------------------------
__global__ void __launch_bounds__(256) gnn_spmm_scatter(
    const int* __restrict__ rows, const int* __restrict__ cols,
    const float* __restrict__ vals, const float* __restrict__ x,
    float* __restrict__ acc, int nnz)
{
    const int wave = threadIdx.x >> 5;
    const int lane = threadIdx.x & 31;
    const int e = blockIdx.x * 8 + wave;
    if (e >= nnz) return;

    // Stream-prefetch the edge lists ahead (global_prefetch_b8, speculative).
    __builtin_prefetch(rows + e + 4096, 0, 1);
    __builtin_prefetch(cols + e + 4096, 0, 1);
    __builtin_prefetch(vals + e + 4096, 0, 1);

    const int   r = __builtin_amdgcn_readfirstlane(rows[e]);
    const int   c = __builtin_amdgcn_readfirstlane(cols[e]);
    const float v = vals[e];

    const float2 f = ((const float2*)(x + (size_t)c * DIM))[lane];
    float* dst = acc + (size_t)r * DIM + lane * 2;
    atomicAdd(dst + 0, v * f.x);
    atomicAdd(dst + 1, v * f.y);
}

// ---------------------------------------------------------------------------
// Combine: cur = 0.5*acc ; sum += cur ; acc = 0.   float4-vectorized.
// ---------------------------------------------------------------------------
__global__ void __launch_bounds__(256) gnn_combine(
    float4* __restrict__ u_cur, float4* __restrict__ i_cur,
    float4* __restrict__ u_acc, float4* __restrict__ i_acc,
    float4* __restrict__ u_sum, float4* __restrict__ i_sum,
    int nu4, int ni4)
{
    int idx = blockIdx.x * blockDim.x + threadIdx.x;
    float4 *acc, *cur, *sum;
    int j;
    if (idx < nu4) { acc = u_acc; cur = u_cur; sum = u_sum; j = idx; }
    else {
        j = idx - nu4;
        if (j >= ni4) return;
        acc = i_acc; cur = i_cur; sum = i_sum;
    }
    float4 a = acc[j];
    a.x *= 0.5f; a.y *= 0.5f; a.z *= 0.5f; a.w *= 0.5f;
    cur[j] = a;
    float4 s = sum[j];
    s.x += a.x; s.y += a.y; s.z += a.z; s.w += a.w;
    sum[j] = s;
    acc[j] = make_float4(0.f, 0.f, 0.f, 0.f);
}

// ---------------------------------------------------------------------------
// Finalize: data *= scale, computed as D = (scale * I16) x X per 16x16 f32
// tile via chained V_WMMA_F32_16X16X4_F32 (exact in f32).
//
// ISA layouts used (wave32):
//   A (16x4 f32, 2 VGPRs):  M = lane&15; VGPR0 holds K = (lane<16 ? 0 : 2),
//                           VGPR1 holds K = (lane<16 ? 1 : 3).
//   B (4x16 f32, 2 VGPRs):  N = lane&15; VGPR0 row K = (lane<16 ? 0 : 2),
//                           VGPR1 row K = (lane<16 ? 1 : 3).
//   C/D (16x16 f32, 8 VGPRs): VGPR v -> M = v (lanes 0-15) / v+8 (lanes 16-31),
//                           N = lane&15.
// One wave handles a 16-row x 64-col block (4 column tiles x 4 K-chunks
// = 16 WMMAs). EXEC is all-ones: the tile-bound branch is wave-uniform.
// ---------------------------------------------------------------------------
__global__ void __launch_bounds__(256) gnn_wmma_finalize(
    float* __restrict__ data, int ntiles, float scale)
{
    const int wave = threadIdx.x >> 5;
    const int lane = threadIdx.x & 31;
    const int tile = blockIdx.x * 8 + wave;
    if (tile >= ntiles) return;

    const int half = lane >> 4;   // 0: lanes 0-15, 1: lanes 16-31
    const int n    = lane & 15;   // N index (and M index for A)
    const int R    = tile * 16;   // base row of this tile block

    // A chunks: slice k of (scale * I16): A_k[m, kk] = scale iff m == 4k+kk.
    v2f a[4];
#pragma unroll
    for (int k = 0; k < 4; ++k) {
        const int k0 = 4 * k + half * 2;
        a[k].x = (n == k0)     ? scale : 0.f;
        a[k].y = (n == k0 + 1) ? scale : 0.f;
    }

#pragma unroll
    for (int j = 0; j < 4; ++j) {      // 4 column tiles covering 64 dims
        v8f acc = {};
        const float* col = data + (size_t)R * DIM + j * 16 + n;
#pragma unroll
        for (int k = 0; k < 4; ++k) {  // K sweep over the tile's 16 rows
            const int kr = 4 * k + half * 2;
            v2f b;
            b.x = col[(size_t)(kr)     * DIM];
            b.y = col[(size_t)(kr + 1) * DIM];
            // 8 args: (neg_a, A, neg_b, B, c_mod, C, reuse_a, reuse_b)
            acc = __builtin_amdgcn_wmma_f32_16x16x4_f32(
                false, a[k], false, b, (short)0, acc, false, false);
        }
        float* out = data + (size_t)(R + half * 8) * DIM + j * 16 + n;
#pragma unroll
        for (int v = 0; v < 8; ++v)
            out[(size_t)v * DIM] = acc[v];
    }
}

// ---------------------------------------------------------------------------
// Launch
// ---------------------------------------------------------------------------
extern "C" void kernel_launch(void* const* d_in, const int* in_sizes, int n_in,
                              void* d_out, int out_size, void* d_ws, size_t ws_size,
                              hipStream_t stream)
{
    const float* ue      = (const float*)d_in[0];
    const float* ie      = (const float*)d_in[1];
    const float* uu_val  = (const float*)d_in[2];
    const float* ui_val  = (const float*)d_in[3];
    const float* ii_val  = (const float*)d_in[4];
    const int*   uu_rows = (const int*)d_in[5];
    const int*   uu_cols = (const int*)d_in[6];
    const int*   ui_rows = (const int*)d_in[7];
    const int*   ui_cols = (const int*)d_in[8];
    const int*   ii_rows = (const int*)d_in[9];
    const int*   ii_cols = (const int*)d_in[10];

    const int NU = in_sizes[0];   // user_num * 64  = 6,400,000
    const int NI = in_sizes[1];   // item_num * 64  = 3,200,000
    const int nnz_uu = in_sizes[2];
    const int nnz_ui = in_sizes[3];
    const int nnz_ii = in_sizes[4];

    // Workspace: u_cur | i_cur | u_acc | i_acc  (= 76.8 MB)
    float* u_cur = (float*)d_ws;
    float* i_cur = u_cur + NU;
    float* u_acc = i_cur + NI;
    float* i_acc = u_acc + NU;

    // Running sums live directly in d_out (u_sum then i_sum, flat).
    float* u_sum = (float*)d_out;
    float* i_sum = u_sum + NU;

    const int nu4 = NU / 4, ni4 = NI / 4;
    const int n4  = nu4 + ni4;
    dim3 blk(256);

    gnn_init<<<(n4 + 255) / 256, blk, 0, stream>>>(
        (const float4*)ue, (const float4*)ie,
        (float4*)u_cur, (float4*)i_cur,
        (float4*)u_acc, (float4*)i_acc,
        (float4*)u_sum, (float4*)i_sum, nu4, ni4);

    for (int layer = 0; layer < 2; ++layer) {
        // u_acc += UI @ i_cur ; i_acc += UI^T @ u_cur ;
        // u_acc += UU @ u_cur ; i_acc += II @ i_cur
        gnn_spmm_scatter<<<(nnz_ui + 7) / 8, blk, 0, stream>>>(
            ui_rows, ui_cols, ui_val, i_cur, u_acc, nnz_ui);
        gnn_spmm_scatter<<<(nnz_ui + 7) / 8, blk, 0, stream>>>(
            ui_cols, ui_rows, ui_val, u_cur, i_acc, nnz_ui);
        gnn_spmm_scatter<<<(nnz_uu + 7) / 8, blk, 0, stream>>>(
            uu_rows, uu_cols, uu_val, u_cur, u_acc, nnz_uu);
        gnn_spmm_scatter<<<(nnz_ii + 7) / 8, blk, 0, stream>>>(
            ii_rows, ii_cols, ii_val, i_cur, i_acc, nnz_ii);

        gnn_combine<<<(n4 + 255) / 256, blk, 0, stream>>>(
            (float4*)u_cur, (float4*)i_cur,
            (float4*)u_acc, (float4*)i_acc,
            (float4*)u_sum, (float4*)i_sum, nu4, ni4);
    }

    // Final mean over (N_LAYERS+1) = divide by 3, as exact f32 WMMA tiles.
    // (USER_NUM + ITEM_NUM) = 150000 rows, divisible by 16 -> 9375 tiles.
    const int total_rows = (NU + NI) / DIM;
    const int ntiles = total_rows / 16;
    gnn_wmma_finalize<<<(ntiles + 7) / 8, blk, 0, stream>>>(
        (float*)d_out, ntiles, 1.0f / 3.0f);
}